// LCN_77773267796428
// MI455X (gfx1250) — compile-verified
//
#include <hip/hip_runtime.h>
#include <math.h>

#define J 17
#define C 512
#define JC 8704          // J*C
#define BATCH 64
#define TC 64
#define TE 256
#define EPS 1e-5f
#define NSPLIT 4         // split-K factor over ji blocks

typedef __attribute__((ext_vector_type(2))) float v2f;
typedef __attribute__((ext_vector_type(8))) float v8f;

// ---------------------------------------------------------------------------
// 1. timestep embedding: emb[n, 0:32]=sin(t*f_i), emb[n,32:64]=cos(t*f_i)
// ---------------------------------------------------------------------------
__global__ void embed_kernel(const int* __restrict__ t, float* __restrict__ emb) {
    int idx = blockIdx.x * blockDim.x + threadIdx.x;   // 64*64
    if (idx >= BATCH * TC) return;
    int n = idx >> 6;
    int h = idx & 63;
    int i = h & 31;
    float freq = __expf((float)i * (-logf(10000.0f) / 31.0f));
    float arg  = (float)t[n] * freq;
    emb[idx] = (h < 32) ? sinf(arg) : cosf(arg);
}

// ---------------------------------------------------------------------------
// 2. small dense layer + SiLU  (time MLP): out[n,o] = silu(in[n,:]@W[o,:]+b[o])
// ---------------------------------------------------------------------------
__global__ void dense_silu_kernel(const float* __restrict__ in,
                                  const float* __restrict__ W,
                                  const float* __restrict__ b,
                                  float* __restrict__ out,
                                  int N, int K, int O) {
    int idx = blockIdx.x * blockDim.x + threadIdx.x;
    if (idx >= N * O) return;
    int n = idx / O, o = idx - n * O;
    const float* xr = in + (size_t)n * K;
    const float* wr = W + (size_t)o * K;
    float s = b[o];
    for (int k = 0; k < K; ++k) s = fmaf(xr[k], wr[k], s);
    s = s / (1.0f + __expf(-s));
    out[idx] = s;
}

// ---------------------------------------------------------------------------
// 3. start layer: raw[n,o] = b[o] + temb[n, o%C] + sum_{masked ji} xin.W
// ---------------------------------------------------------------------------
__global__ void start_kernel(const float* __restrict__ xin,
                             const float* __restrict__ W,
                             const float* __restrict__ b,
                             const float* __restrict__ temb,
                             const float* __restrict__ mask,
                             float* __restrict__ Y) {
    int idx = blockIdx.x * blockDim.x + threadIdx.x;   // 64*8704
    if (idx >= BATCH * JC) return;
    int n = idx / JC, o = idx - n * JC;
    int jo = o >> 9, c = o & (C - 1);
    float s = b[o] + temb[n * C + c];
    const float* wr = W + (size_t)o * (J * 3);
    const float* xr = xin + n * (J * 3);
    for (int ji = 0; ji < J; ++ji) {
        if (mask[jo * J + ji] != 0.0f) {
            int k = ji * 3;
            s = fmaf(wr[k], xr[k], s);
            s = fmaf(wr[k + 1], xr[k + 1], s);
            s = fmaf(wr[k + 2], xr[k + 2], s);
        }
    }
    Y[idx] = s;
}

// ---------------------------------------------------------------------------
// 4. BatchNorm stats: per-channel mean / rstd over (N=64, J=17) = 1088 samples
// ---------------------------------------------------------------------------
__global__ void bn_stats_kernel(const float* __restrict__ X,
                                float* __restrict__ mean,
                                float* __restrict__ rstd) {
    int c = blockIdx.x;                                // 0..511
    float s = 0.0f, s2 = 0.0f;
    for (int idx = threadIdx.x; idx < BATCH * J; idx += blockDim.x) {
        int n = idx / J, j = idx - n * J;
        float v = X[(size_t)n * JC + j * C + c];
        s += v;
        s2 = fmaf(v, v, s2);
    }
    __shared__ float sh[256], sh2[256];
    sh[threadIdx.x] = s; sh2[threadIdx.x] = s2;
    __syncthreads();
    for (int off = 128; off > 0; off >>= 1) {
        if ((int)threadIdx.x < off) {
            sh[threadIdx.x]  += sh[threadIdx.x + off];
            sh2[threadIdx.x] += sh2[threadIdx.x + off];
        }
        __syncthreads();
    }
    if (threadIdx.x == 0) {
        float inv = 1.0f / (float)(BATCH * J);
        float mu  = sh[0] * inv;
        float var = sh2[0] * inv - mu * mu;
        mean[c] = mu;
        rstd[c] = rsqrtf(var + EPS);
    }
}

// ---------------------------------------------------------------------------
// 5. BN apply + SiLU (+ optional residual add)
// ---------------------------------------------------------------------------
__global__ void bn_apply_kernel(const float* __restrict__ X,
                                const float* __restrict__ mean,
                                const float* __restrict__ rstd,
                                const float* __restrict__ gamma,
                                const float* __restrict__ beta,
                                const float* __restrict__ res,  // nullable
                                float* __restrict__ Y) {
    int idx = blockIdx.x * blockDim.x + threadIdx.x;   // 64*8704
    if (idx >= BATCH * JC) return;
    int c = idx & (C - 1);
    float v = (X[idx] - mean[c]) * rstd[c] * gamma[c] + beta[c];
    v = v / (1.0f + __expf(-v));                       // SiLU
    if (res) v += res[idx];
    Y[idx] = v;
}

// ---------------------------------------------------------------------------
// 6. masked block-GEMM, split-K over ji blocks for occupancy.
//    grid = (68, NSPLIT); block = 256 (8 waves). Wave w owns output strip
//    [bx*128 + w*16, +16); split s owns ji in {s, s+NSPLIT, ...} (mask-skipped,
//    wave-uniform branch -> EXEC all-ones around WMMA). Partial sums written
//    to part[s] (full strip written even if all ji masked -> no stale data).
//    V_WMMA_F32_16X16X4_F32 layouts:
//      A (16x4):  lanes 0-15 hold M=lane,K={0,1}; lanes 16-31 M,K={2,3}
//      B (4x16):  lanes 0-15 hold N=lane,K={0,1}; lanes 16-31 N,K={2,3}
//      D (16x16): lane -> N=lane&15, vgpr r -> M = r + 8*(lane>=16)
// ---------------------------------------------------------------------------
__global__ __launch_bounds__(256) void gemm_split_kernel(
    const float* __restrict__ X,
    const float* __restrict__ W,
    const float* __restrict__ mask,
    float* __restrict__ part) {
    const int wave = threadIdx.x >> 5;
    const int lane = threadIdx.x & 31;
    const int lrow = lane & 15;
    const int half = lane >> 4;
    const int nbase = blockIdx.x * 128 + wave * 16;    // output-channel strip
    const int jo = nbase >> 9;
    const int s = blockIdx.y;

    v8f acc0 = {}, acc1 = {}, acc2 = {}, acc3 = {};

    const float* wrow = W + (size_t)(nbase + lrow) * JC + 2 * half;
    const float* x0 = X + (size_t)lrow * JC + 2 * half;
    const float* x1 = x0 + 16 * JC;
    const float* x2 = x0 + 32 * JC;
    const float* x3 = x0 + 48 * JC;

    for (int ji = s; ji < J; ji += NSPLIT) {
        if (mask[jo * J + ji] == 0.0f) continue;       // wave-uniform skip
        const int base = ji * C;
        __builtin_prefetch(wrow + base, 0, 1);         // global_prefetch_b8
        __builtin_prefetch(wrow + base + 256, 0, 1);
#pragma unroll 4
        for (int kk = 0; kk < C; kk += 4) {
            const int i = base + kk;
            v2f bw = *(const v2f*)(wrow + i);
            v2f a0 = *(const v2f*)(x0 + i);
            v2f a1 = *(const v2f*)(x1 + i);
            v2f a2 = *(const v2f*)(x2 + i);
            v2f a3 = *(const v2f*)(x3 + i);
            acc0 = __builtin_amdgcn_wmma_f32_16x16x4_f32(
                false, a0, false, bw, (short)0, acc0, false, false);
            acc1 = __builtin_amdgcn_wmma_f32_16x16x4_f32(
                false, a1, false, bw, (short)0, acc1, false, false);
            acc2 = __builtin_amdgcn_wmma_f32_16x16x4_f32(
                false, a2, false, bw, (short)0, acc2, false, false);
            acc3 = __builtin_amdgcn_wmma_f32_16x16x4_f32(
                false, a3, false, bw, (short)0, acc3, false, false);
        }
    }

    // epilogue: D tile column n = lane&15; row m = r + 8*half (+16 per m-tile)
    float* yc = part + (size_t)s * (BATCH * JC)
              + (size_t)(half * 8) * JC + nbase + lrow;
#pragma unroll
    for (int r = 0; r < 8; ++r) {
        yc[(size_t)r * JC]        = acc0[r];
        yc[(size_t)(r + 16) * JC] = acc1[r];
        yc[(size_t)(r + 32) * JC] = acc2[r];
        yc[(size_t)(r + 48) * JC] = acc3[r];
    }
}

// ---------------------------------------------------------------------------
// 6b. deterministic split-K reduction:  raw = sum(partials) + bias
// ---------------------------------------------------------------------------
__global__ void reduce_bias_kernel(const float* __restrict__ part,
                                   const float* __restrict__ bias,
                                   float* __restrict__ Y) {
    int idx = blockIdx.x * blockDim.x + threadIdx.x;   // 64*8704
    if (idx >= BATCH * JC) return;
    int o = idx % JC;
    const int TOT = BATCH * JC;
    float s = bias[o];
    s += part[idx];
    s += part[idx + TOT];
    s += part[idx + 2 * TOT];
    s += part[idx + 3 * TOT];
    Y[idx] = s;
}

// ---------------------------------------------------------------------------
// 7. final layer: out[n, o<51] = b[o] + sum_{masked ji} act[n,ji*C:]·Wf[o,...]
// ---------------------------------------------------------------------------
__global__ void final_kernel(const float* __restrict__ Xa,
                             const float* __restrict__ W,
                             const float* __restrict__ b,
                             const float* __restrict__ mask,
                             float* __restrict__ out) {
    int n = blockIdx.x;
    int o = threadIdx.x;
    if (o >= J * 3) return;
    int jo = o / 3;
    float s = b[o];
    for (int ji = 0; ji < J; ++ji) {
        if (mask[jo * J + ji] == 0.0f) continue;
        const float* xr = Xa + (size_t)n * JC + ji * C;
        const float* wr = W + (size_t)o * JC + ji * C;
        for (int c = 0; c < C; ++c) s = fmaf(xr[c], wr[c], s);
    }
    out[n * (J * 3) + o] = s;
}

// ---------------------------------------------------------------------------
// launcher
// ---------------------------------------------------------------------------
extern "C" void kernel_launch(void* const* d_in, const int* in_sizes, int n_in,
                              void* d_out, int out_size, void* d_ws, size_t ws_size,
                              hipStream_t stream) {
    const float* xin     = (const float*)d_in[0];
    const int*   time_t_ = (const int*)  d_in[1];
    const float* mask    = (const float*)d_in[2];
    const float* W_start = (const float*)d_in[3];
    const float* b_start = (const float*)d_in[4];
    const float* Wb      = (const float*)d_in[5];
    const float* bb      = (const float*)d_in[6];
    const float* W_final = (const float*)d_in[7];
    const float* b_final = (const float*)d_in[8];
    const float* Wt0 = (const float*)d_in[9];
    const float* bt0 = (const float*)d_in[10];
    const float* Wt1 = (const float*)d_in[11];
    const float* bt1 = (const float*)d_in[12];
    const float* Wt2 = (const float*)d_in[13];
    const float* bt2 = (const float*)d_in[14];
    const float* gamma = (const float*)d_in[15];
    const float* beta  = (const float*)d_in[16];
    float* out = (float*)d_out;

    // workspace layout (floats); total ~3.97M floats = 15.9 MB
    float* ws   = (float*)d_ws;
    float* emb  = ws;                       //  64*64    =   4096
    float* t0   = ws + 4096;                //  64*256   =  16384
    float* t1   = ws + 20480;               //  64*256   =  16384
    float* temb = ws + 36864;               //  64*512   =  32768
    float* raw  = ws + 69632;               //  64*8704  = 557056
    float* bP   = ws + 626688;              //  557056
    float* bQ   = ws + 1183744;             //  557056
    float* mean = ws + 1740800;             //  512
    float* rstd = ws + 1741312;             //  512
    float* part = ws + 1741824;             //  4*557056 = 2228224

    const size_t NW = (size_t)JC * (size_t)JC;   // elems per block-layer weight
    const int TOT = BATCH * JC;                  // 557056
    const int GB_E = (TOT + 255) / 256;          // elementwise grid = 2176
    const dim3 GGEMM(68, NSPLIT);

    // ---- time-embedding MLP ----
    embed_kernel<<<(BATCH * TC + 255) / 256, 256, 0, stream>>>(time_t_, emb);
    dense_silu_kernel<<<(BATCH * TE + 255) / 256, 256, 0, stream>>>(emb, Wt0, bt0, t0, BATCH, TC, TE);
    dense_silu_kernel<<<(BATCH * TE + 255) / 256, 256, 0, stream>>>(t0, Wt1, bt1, t1, BATCH, TE, TE);
    dense_silu_kernel<<<(BATCH * C + 255) / 256, 256, 0, stream>>>(t1, Wt2, bt2, temb, BATCH, TE, C);

    // ---- start layer + BN + SiLU ----
    start_kernel<<<GB_E, 256, 0, stream>>>(xin, W_start, b_start, temb, mask, raw);
    bn_stats_kernel<<<C, 256, 0, stream>>>(raw, mean, rstd);
    bn_apply_kernel<<<GB_E, 256, 0, stream>>>(raw, mean, rstd, gamma, beta, nullptr, bP);

    // ---- block 0 (residual source: bP) ----
    gemm_split_kernel<<<GGEMM, 256, 0, stream>>>(bP, Wb + 0 * NW, mask, part);
    reduce_bias_kernel<<<GB_E, 256, 0, stream>>>(part, bb + 0 * JC, raw);
    bn_stats_kernel<<<C, 256, 0, stream>>>(raw, mean, rstd);
    bn_apply_kernel<<<GB_E, 256, 0, stream>>>(raw, mean, rstd, gamma, beta, nullptr, bQ);

    gemm_split_kernel<<<GGEMM, 256, 0, stream>>>(bQ, Wb + 1 * NW, mask, part);
    reduce_bias_kernel<<<GB_E, 256, 0, stream>>>(part, bb + 1 * JC, raw);
    bn_stats_kernel<<<C, 256, 0, stream>>>(raw, mean, rstd);
    bn_apply_kernel<<<GB_E, 256, 0, stream>>>(raw, mean, rstd, gamma, beta, bP, bQ);

    // ---- block 1 (residual source: bQ) ----
    gemm_split_kernel<<<GGEMM, 256, 0, stream>>>(bQ, Wb + 2 * NW, mask, part);
    reduce_bias_kernel<<<GB_E, 256, 0, stream>>>(part, bb + 2 * JC, raw);
    bn_stats_kernel<<<C, 256, 0, stream>>>(raw, mean, rstd);
    bn_apply_kernel<<<GB_E, 256, 0, stream>>>(raw, mean, rstd, gamma, beta, nullptr, bP);

    gemm_split_kernel<<<GGEMM, 256, 0, stream>>>(bP, Wb + 3 * NW, mask, part);
    reduce_bias_kernel<<<GB_E, 256, 0, stream>>>(part, bb + 3 * JC, raw);
    bn_stats_kernel<<<C, 256, 0, stream>>>(raw, mean, rstd);
    bn_apply_kernel<<<GB_E, 256, 0, stream>>>(raw, mean, rstd, gamma, beta, bQ, bP);

    // ---- final projection ----
    final_kernel<<<BATCH, 64, 0, stream>>>(bP, W_final, b_final, mask, out);
}